// QueryKeyMul_27152783245419
// MI455X (gfx1250) — compile-verified
//
#include <hip/hip_runtime.h>

#define HDIM    128
#define TILE_N  128      // keys per workgroup K-tile
#define SUBT    8        // 8 column sub-tiles of 16
#define ROWTILE 64       // 4 waves x 16 rows per workgroup
#define LDSS    132      // padded LDS row stride (floats) -> conflict-free b64
#define MAXB    4096     // max recoverable batches (actual: 64)

typedef __attribute__((ext_vector_type(2))) float v2f;
typedef __attribute__((ext_vector_type(4))) float v4f;
typedef __attribute__((ext_vector_type(8))) float v8f;
typedef __attribute__((ext_vector_type(4))) int   v4i;

typedef __attribute__((address_space(1))) v4i gv4i;   // global v4i
typedef __attribute__((address_space(3))) v4i lv4i;   // LDS v4i

#if __has_builtin(__builtin_amdgcn_global_load_async_to_lds_b128)
#define USE_ASYNC_LDS 1
#else
#define USE_ASYNC_LDS 0
#endif

__device__ __forceinline__ void wait_async_then_barrier() {
#if USE_ASYNC_LDS
#if __has_builtin(__builtin_amdgcn_s_wait_asynccnt)
  __builtin_amdgcn_s_wait_asynccnt(0);
#else
  asm volatile("s_wait_asynccnt 0x0" ::: "memory");
#endif
#endif
  __syncthreads();
}

// ---------------------------------------------------------------------------
// Pass 1: one linear sweep over the P pair indices. Each query's pairs are a
// contiguous run, so a run boundary at p gives row_off[q]=p, row_kstart[q]=k.
// ---------------------------------------------------------------------------
__global__ void rowscan_kernel(const int* __restrict__ qidx,
                               const int* __restrict__ kidx,
                               int P, int Tq,
                               int* __restrict__ row_off,
                               int* __restrict__ row_kstart) {
  int p = blockIdx.x * blockDim.x + threadIdx.x;
  int stride = gridDim.x * blockDim.x;
  for (; p < P; p += stride) {
    int q = qidx[p];
    if (p == 0 || qidx[p - 1] != q) {
      row_off[q]    = p;
      row_kstart[q] = kidx[p];
    }
    if (p == 0) row_off[Tq] = P;
  }
}

// ---------------------------------------------------------------------------
// Pass 2: serial scan over Tq (~2K) rows: a batch starts where row_kstart
// changes. Build per-batch records + tile prefix sum (64-row x 128-col tiles).
// ---------------------------------------------------------------------------
__global__ void buildbatches_kernel(int Tq,
                                    const int* __restrict__ row_off,
                                    const int* __restrict__ row_kstart,
                                    int* __restrict__ meta,
                                    int* __restrict__ brow0,
                                    int* __restrict__ bnrows,
                                    int* __restrict__ bkstart,
                                    int* __restrict__ bnk,
                                    int* __restrict__ bpoff,
                                    int* __restrict__ btoff) {
  if (blockIdx.x != 0 || threadIdx.x != 0) return;
  int nb = 0;
  for (int r = 0; r < Tq; ++r) {
    if (r == 0 || row_kstart[r] != row_kstart[r - 1]) {
      if (nb < MAXB) brow0[nb++] = r;
    }
  }
  int toff = 0;
  for (int b = 0; b < nb; ++b) {
    int r0    = brow0[b];
    int rend  = (b + 1 < nb) ? brow0[b + 1] : Tq;
    int nrows = rend - r0;
    int nk    = row_off[r0 + 1] - row_off[r0];
    bnrows[b]  = nrows;
    bkstart[b] = row_kstart[r0];
    bnk[b]     = nk;
    bpoff[b]   = row_off[r0];
    btoff[b]   = toff;
    toff += ((nrows + ROWTILE - 1) / ROWTILE) * ((nk + TILE_N - 1) / TILE_N);
  }
  btoff[nb] = toff;
  meta[0] = nb;
  meta[1] = toff;
}

// ---------------------------------------------------------------------------
// Pass 3: 128-thread (4-wave) workgroup per 64x128 output tile.
//   stage: K tile (128 cols x 128 h) -> LDS via async global->LDS (ASYNCcnt)
//   compute: each wave does a 16-row block; f32 WMMA 16x16x4, B frags from LDS
// ---------------------------------------------------------------------------
__global__ __launch_bounds__(128)
void qk_wmma_kernel(const float* __restrict__ Q,
                    const float* __restrict__ K,
                    float* __restrict__ out,
                    const int* __restrict__ meta,
                    const int* __restrict__ brow0,
                    const int* __restrict__ bnrows,
                    const int* __restrict__ bkstart,
                    const int* __restrict__ bnk,
                    const int* __restrict__ bpoff,
                    const int* __restrict__ btoff) {
  __shared__ float lds_k[TILE_N * LDSS];   // ~66 KB of the 320 KB WGP LDS

  const int tid  = threadIdx.x;
  const int lane = tid & 31;
  const int wv   = tid >> 5;               // wave id 0..3 -> row block
  const int nb   = meta[0];
  const int T    = meta[1];

  for (int t = blockIdx.x; t < T; t += gridDim.x) {
    // locate batch for this tile (nb is small; linear scan over prefix sums)
    int b = 0;
    while (b + 1 < nb && btoff[b + 1] <= t) ++b;
    const int lt      = t - btoff[b];
    const int nk      = bnk[b];
    const int ncols_t = (nk + TILE_N - 1) / TILE_N;
    const int ti      = lt / ncols_t;
    const int tj      = lt % ncols_t;
    const int row0    = brow0[b];
    const int nrows   = bnrows[b];
    const int kstart  = bkstart[b];
    const int poff    = bpoff[b];

    const int r0 = row0 + ROWTILE * ti + 16 * wv;  // this wave's 16-row block
    const int c0 = TILE_N * tj;                    // first local key column

    // ---- stage K tile into LDS: 4096 b128 units, coalesced per wave ----
    for (int u = tid; u < TILE_N * 32; u += 128) {
      int col  = u >> 5;                 // 0..127
      int hu   = u & 31;                 // 16B chunk within the 512B key row
      int colc = c0 + col;
      if (colc > nk - 1) colc = nk - 1;
      const float* g = K + (size_t)(kstart + colc) * HDIM + hu * 4;
      float*       l = &lds_k[col * LDSS + hu * 4];
#if USE_ASYNC_LDS
      float* gnc = const_cast<float*>(g);
      __builtin_amdgcn_global_load_async_to_lds_b128(
          (gv4i*)gnc, (lv4i*)l, 0, 0);
#else
      *(v4f*)l = *(const v4f*)g;         // global_load_b128 + ds_store_b128
#endif
    }
    wait_async_then_barrier();

    // ---- A fragment address: lane l holds M = l&15, K = (l>>4)*2+{0,1} ----
    int arow   = r0 + (lane & 15);
    int arow_c = arow < (row0 + nrows - 1) ? arow : (row0 + nrows - 1);
    const float* aptr = Q + (size_t)arow_c * HDIM + ((lane >> 4) * 2);

    // ---- B fragments from LDS: lane l holds N = l&15, K = (l>>4)*2+{0,1} --
    const float* bptr[SUBT];
#pragma unroll
    for (int j = 0; j < SUBT; ++j) {
      bptr[j] = &lds_k[(j * 16 + (lane & 15)) * LDSS + ((lane >> 4) * 2)];
    }

    v8f acc[SUBT] = {};

#pragma unroll 4
    for (int kk = 0; kk < HDIM; kk += 4) {
      v2f a = *(const v2f*)(aptr + kk);
#pragma unroll
      for (int j = 0; j < SUBT; ++j) {
        v2f bf = *(const v2f*)(bptr[j] + kk);   // ds_load_b64, conflict-free
        acc[j] = __builtin_amdgcn_wmma_f32_16x16x4_f32(
            false, a, false, bf, (short)0, acc[j], false, false);
      }
    }

    // ---- store: VGPR v, lane l -> m = v + (l>>4)*8, n = l&15 ----
    const int m_base = (lane >> 4) * 8;
    const int n      = lane & 15;
#pragma unroll
    for (int j = 0; j < SUBT; ++j) {
#pragma unroll
      for (int v = 0; v < 8; ++v) {
        int rr = r0 + m_base + v;
        int cc = c0 + j * 16 + n;
        if (rr < row0 + nrows && cc < nk) {
          out[(size_t)poff + (size_t)(rr - row0) * nk + cc] = acc[j][v];
        }
      }
    }
    __syncthreads();   // protect LDS before next tile's staging
  }
}

// ---------------------------------------------------------------------------
extern "C" void kernel_launch(void* const* d_in, const int* in_sizes, int n_in,
                              void* d_out, int out_size, void* d_ws, size_t ws_size,
                              hipStream_t stream) {
  const float* Q    = (const float*)d_in[0];
  const float* K    = (const float*)d_in[1];
  const int*   qidx = (const int*)d_in[2];
  const int*   kidx = (const int*)d_in[3];
  float*       out  = (float*)d_out;

  const int P  = in_sizes[2];
  const int Tq = in_sizes[0] / HDIM;

  // workspace layout (ints)
  int* w          = (int*)d_ws;
  int* meta       = w;                    // [0]=nb, [1]=total tiles
  int* brow0      = w + 2;                // MAXB
  int* bnrows     = brow0  + MAXB;        // MAXB
  int* bkstart    = bnrows + MAXB;        // MAXB
  int* bnk        = bkstart + MAXB;       // MAXB
  int* bpoff      = bnk    + MAXB;        // MAXB
  int* btoff      = bpoff  + MAXB;        // MAXB + 1
  int* row_off    = btoff  + MAXB + 1;    // Tq + 1
  int* row_kstart = row_off + Tq + 1;     // Tq

  int blocks1 = (P + 255) / 256;
  if (blocks1 > 4096) blocks1 = 4096;
  rowscan_kernel<<<blocks1, 256, 0, stream>>>(qidx, kidx, P, Tq,
                                              row_off, row_kstart);

  buildbatches_kernel<<<1, 1, 0, stream>>>(Tq, row_off, row_kstart, meta,
                                           brow0, bnrows, bkstart, bnk,
                                           bpoff, btoff);

  qk_wmma_kernel<<<1024, 128, 0, stream>>>(Q, K, out, meta, brow0, bnrows,
                                           bkstart, bnk, bpoff, btoff);
}